// GraphSAGE_27358941676271
// MI455X (gfx1250) — compile-verified
//
#include <hip/hip_runtime.h>

typedef __attribute__((ext_vector_type(2))) float v2f;
typedef __attribute__((ext_vector_type(8))) float v8f;

#define ROWS   16          // output rows per block (M tile)
#define FDIM   256         // feature dim F
#define KNB    32          // neighbors per node
#define HCOLS  512         // concat width 2F
#define HPAD   4           // LDS pad (floats) -> stride 516, bank-conflict-free A reads
#define HSTR   (HCOLS + HPAD)

__global__ __launch_bounds__(256)
void sage_fused_kernel(const float* __restrict__ x_self,
                       const float* __restrict__ feats,
                       const int*   __restrict__ neigh_idx,
                       const float* __restrict__ W,     // (512, 256) row-major
                       const float* __restrict__ bias,  // (256,)
                       float* __restrict__ out)         // (B, 256)
{
    __shared__ float h[ROWS * HSTR];       // [16][516] concat(x_self, mean_neigh)
    __shared__ int   sidx[ROWS * KNB];     // 512 neighbor indices

    const int t    = threadIdx.x;
    const int row0 = blockIdx.x * ROWS;

    // ---- stage neighbor indices for this block's 16 rows ----
    for (int i = t; i < ROWS * KNB; i += 256)
        sidx[i] = neigh_idx[row0 * KNB + i];

    // ---- x_self tile -> h[:, 0:256] (coalesced b32) ----
    #pragma unroll 4
    for (int r = 0; r < ROWS; ++r)
        h[r * HSTR + t] = x_self[(size_t)(row0 + r) * FDIM + t];

    __syncthreads();

    // ---- gather + mean -> h[:, 256:512] ----
    // 256 threads = 64 column-groups (float4) x 4 row-groups; each wave issues
    // contiguous 512B loads from a (random) 1KB feats row.
    {
        const int cg = t & 63;      // column group: cols [4cg, 4cg+3]
        const int rg = t >> 6;      // row group 0..3 -> rows rg*4 .. rg*4+3
        const float s = 1.0f / (float)KNB;
        for (int rr = 0; rr < 4; ++rr) {
            const int r = rg * 4 + rr;
            float ax = 0.f, ay = 0.f, az = 0.f, aw = 0.f;
            const float* colbase = feats + cg * 4;
            #pragma unroll 4
            for (int k = 0; k < KNB; ++k) {
                const int idx = sidx[r * KNB + k];
                const float4 v = *reinterpret_cast<const float4*>(colbase + (size_t)idx * FDIM);
                ax += v.x; ay += v.y; az += v.z; aw += v.w;
            }
            float* dst = &h[r * HSTR + FDIM + cg * 4];
            dst[0] = ax * s; dst[1] = ay * s; dst[2] = az * s; dst[3] = aw * s;
        }
    }
    __syncthreads();

    // ---- fp32 WMMA GEMM: out[row0:row0+16, :] = h @ W + b ----
    // 8 waves x 2 n-tiles each = 16 tiles of 16x16 covering all 256 output cols.
    const int wave = t >> 5;
    const int lane = t & 31;
    const int half = lane >> 4;     // 0: lanes 0-15, 1: lanes 16-31
    const int l16  = lane & 15;

    for (int nt = 0; nt < 2; ++nt) {
        const int n0 = (wave * 2 + nt) * 16;

        // C/D layout: lane -> N = l16, VGPR v -> M = v + 8*half. Bias depends on N only.
        const float bv = bias[n0 + l16];
        v8f acc;
        #pragma unroll
        for (int v = 0; v < 8; ++v) acc[v] = bv;

        // A 16x4 f32: lane(m=l16, half) holds {h[m][k0+2*half], h[m][k0+2*half+1]}
        const float* hA = &h[l16 * HSTR + 2 * half];
        // B 4x16 f32: lane(n=l16, half) holds {W[k0+2*half][n], W[k0+2*half+1][n]}
        const float* wB = &W[(size_t)(2 * half) * FDIM + n0 + l16];

        for (int k0 = 0; k0 < HCOLS; k0 += 4) {
            v2f a;
            a.x = hA[k0];
            a.y = hA[k0 + 1];
            v2f bf;
            bf.x = wB[(size_t)k0 * FDIM];
            bf.y = wB[(size_t)(k0 + 1) * FDIM];
            acc = __builtin_amdgcn_wmma_f32_16x16x4_f32(
                /*neg_a=*/false, a, /*neg_b=*/false, bf,
                /*c_mod=*/(short)0, acc, /*reuse_a=*/false, /*reuse_b=*/false);
        }

        // store D fragment: row = row0 + v + 8*half, col = n0 + l16
        float* obase = out + (size_t)(row0 + 8 * half) * FDIM + n0 + l16;
        #pragma unroll
        for (int v = 0; v < 8; ++v)
            obase[(size_t)v * FDIM] = acc[v];
    }
}

extern "C" void kernel_launch(void* const* d_in, const int* in_sizes, int n_in,
                              void* d_out, int out_size, void* d_ws, size_t ws_size,
                              hipStream_t stream) {
    const float* x_self = (const float*)d_in[0];   // (B, 256)
    const float* feats  = (const float*)d_in[1];   // (N, 256)
    const int*   nidx   = (const int*)d_in[2];     // (B, 32)
    const float* W      = (const float*)d_in[3];   // (512, 256)
    const float* bias   = (const float*)d_in[4];   // (256,)
    float* out = (float*)d_out;                    // (B, 256)

    const int B = in_sizes[0] / FDIM;              // 50000, divisible by 16
    dim3 grid(B / ROWS), block(256);
    sage_fused_kernel<<<grid, block, 0, stream>>>(x_self, feats, nidx, W, bias, out);
}